// LocalSelfAttention_4209067950266
// MI455X (gfx1250) — compile-verified
//
#include <hip/hip_runtime.h>
#include <math.h>

// ---------------------------------------------------------------------------
// LocalSelfAttention for MI455X (gfx1250, wave32, WMMA)
//   x:(4,2048,512) f32, Wqkv:(1536,512), bqkv:(1536), Wout:(512,512), bout:(512)
//   Pipeline: [GEMM qkv] -> [banded attention, window 65] -> [GEMM out]
//   All matrix math on V_WMMA_F32_16X16X4_F32 (fp32-exact).
//   GEMM waves are 2x4 register-blocked; fragments loaded as b128, each
//   feeding TWO WMMAs (custom K-slot assignment: .xy -> {k,k+1|k+4,k+5},
//   .zw -> {k+2,k+3|k+6,k+7}; valid since A/B share the assignment).
// ---------------------------------------------------------------------------

typedef float v2f __attribute__((ext_vector_type(2)));
typedef float v4f __attribute__((ext_vector_type(4)));
typedef float v8f __attribute__((ext_vector_type(8)));

#define D_MODEL   512
#define NUM_HEADS 8
#define DK        64
#define RADIUS    32          // window 65
#define BSZ       4
#define TLEN      2048
#define ROWS      (BSZ * TLEN)        // 8192
#define QKV_COLS  (3 * D_MODEL)       // 1536
#define WKEYS     80                  // 16 queries + 2*RADIUS key span
#define PSTRIDE   84                  // padded LDS row stride (floats)

static __device__ __forceinline__ v8f wmma_f32_k4(v2f a, v2f b, v8f c) {
  // D(16x16,f32) = A(16x4,f32) x B(4x16,f32) + C
  return __builtin_amdgcn_wmma_f32_16x16x4_f32(
      /*neg_a=*/false, a, /*neg_b=*/false, b,
      /*c_mod=*/(short)0, c, /*reuse_a=*/false, /*reuse_b=*/false);
}
static __device__ __forceinline__ v2f lo2(v4f v) { return (v2f){v.x, v.y}; }
static __device__ __forceinline__ v2f hi2(v4f v) { return (v2f){v.z, v.w}; }
static __device__ __forceinline__ v8f zero8() {
  return (v8f){0.f, 0.f, 0.f, 0.f, 0.f, 0.f, 0.f, 0.f};
}

// ---------------------------------------------------------------------------
// C(M,N) = A(M,K) * W(N,K)^T + bias(N)
// One wave computes a 32x64 tile (2 M-subtiles x 4 N-subtiles).
// Per 8-wide K-chunk: 2 A b128 + 4 B b128 loads feed 16 WMMAs.
// C/D layout: VGPR r -> row r + 8*(lane>=16), col lane&15
// ---------------------------------------------------------------------------
__global__ __launch_bounds__(256)
void gemm_wmma_f32(const float* __restrict__ A, int lda,
                   const float* __restrict__ W, int ldw,
                   const float* __restrict__ bias,
                   float* __restrict__ C, int ldc,
                   int tiles_n /* N/64 */, int K) {
  const int lane = threadIdx.x & 31;
  const int wave = threadIdx.x >> 5;
  const int tile = blockIdx.x * 8 + wave;
  const int tm = tile / tiles_n;
  const int tn = tile - tm * tiles_n;
  const int m0 = tm << 5, n0 = tn << 6;
  const int half = lane >> 4;                 // selects K-quad within chunk
  const int l16  = lane & 15;

  const float* arow0 = A + (size_t)(m0 + l16) * lda + 4 * half;
  const float* arow1 = arow0 + (size_t)16 * lda;
  const float* wrow  = W + (size_t)(n0 + l16) * ldw + 4 * half;
  const size_t wstep = (size_t)16 * ldw;

  v8f acc[2][4];
#pragma unroll
  for (int mi = 0; mi < 2; ++mi)
#pragma unroll
    for (int nt = 0; nt < 4; ++nt) acc[mi][nt] = zero8();

  for (int k = 0; k < K; k += 8) {
    v4f a0 = *(const v4f*)(arow0 + k);
    v4f a1 = *(const v4f*)(arow1 + k);
    v4f b[4];
#pragma unroll
    for (int nt = 0; nt < 4; ++nt)
      b[nt] = *(const v4f*)(wrow + (size_t)nt * wstep + k);
#pragma unroll
    for (int nt = 0; nt < 4; ++nt) {
      acc[0][nt] = wmma_f32_k4(lo2(a0), lo2(b[nt]), acc[0][nt]);
      acc[1][nt] = wmma_f32_k4(lo2(a1), lo2(b[nt]), acc[1][nt]);
      acc[0][nt] = wmma_f32_k4(hi2(a0), hi2(b[nt]), acc[0][nt]);
      acc[1][nt] = wmma_f32_k4(hi2(a1), hi2(b[nt]), acc[1][nt]);
    }
  }

#pragma unroll
  for (int nt = 0; nt < 4; ++nt) {
    const float bv = bias[n0 + nt * 16 + l16];
#pragma unroll
    for (int mi = 0; mi < 2; ++mi) {
      float* crow =
          C + (size_t)(m0 + mi * 16 + 8 * half) * ldc + n0 + nt * 16 + l16;
#pragma unroll
      for (int r = 0; r < 8; ++r)
        crow[(size_t)r * ldc] = acc[mi][nt][r] + bv;
    }
  }
}

// ---------------------------------------------------------------------------
// Banded local attention. One wave per (b, h, 16-query block).
// qkv row layout: [q(0..511) | k(512..1023) | v(1024..1535)], head h -> h*64.
// Keys for queries [t0, t0+15] span [t0-32, t0+47] -> 80 local positions.
// scores: 5 WMMA tiles, K-chunks of 8 (b128 Q/K fragment loads).
// softmax: banded mask, shfl_xor row reductions (rows live in 16-lane halves).
// P -> LDS (C-layout -> A-layout transpose), ctx: 4 WMMA tiles over 80 keys.
// ---------------------------------------------------------------------------
__global__ __launch_bounds__(256)
void local_attn_wmma(const float* __restrict__ qkv,
                     float* __restrict__ ctx) {
  __shared__ float p_lds[8][16 * PSTRIDE];    // 43 KB static LDS

  const int lane = threadIdx.x & 31;
  const int wave = threadIdx.x >> 5;
  const int unit = blockIdx.x * 8 + wave;     // 4*8*128 = 4096 units
  const int tb = unit & 127;
  const int h  = (unit >> 7) & 7;
  const int b  = unit >> 10;
  const int t0 = tb << 4;
  const int half = lane >> 4;
  const int l16  = lane & 15;
  const size_t rowbase = (size_t)b * TLEN;

  // ---- scores: S(16x80) = Q(16x64) * Kwin^T ----
  v8f s[5];
#pragma unroll
  for (int nt = 0; nt < 5; ++nt) s[nt] = zero8();

  const float* qptr = qkv + (rowbase + t0 + l16) * QKV_COLS + h * DK + 4 * half;
  const float* kbase[5];
#pragma unroll
  for (int nt = 0; nt < 5; ++nt) {
    int key  = t0 - RADIUS + nt * 16 + l16;
    int keyc = min(max(key, 0), TLEN - 1);
    kbase[nt] = qkv + (rowbase + keyc) * QKV_COLS + D_MODEL + h * DK + 4 * half;
  }
  for (int k0 = 0; k0 < DK; k0 += 8) {
    v4f a = *(const v4f*)(qptr + k0);
#pragma unroll
    for (int nt = 0; nt < 5; ++nt) {
      v4f bf = *(const v4f*)(kbase[nt] + k0);
      s[nt] = wmma_f32_k4(lo2(a), lo2(bf), s[nt]);
      s[nt] = wmma_f32_k4(hi2(a), hi2(bf), s[nt]);
    }
  }

  // ---- scale + banded mask ----
  const float scale = 0.125f;                 // 1/sqrt(64)
#pragma unroll
  for (int nt = 0; nt < 5; ++nt) {
#pragma unroll
    for (int r = 0; r < 8; ++r) {
      int m = r + 8 * half;                   // query row within block
      int n = nt * 16 + l16;                  // local key position
      int key = t0 - RADIUS + n;
      bool valid = (n >= m) && (n <= m + 2 * RADIUS) && (key >= 0) && (key < TLEN);
      s[nt][r] = valid ? s[nt][r] * scale : -3.0e38f;
    }
  }

  // ---- row softmax (rows split across 16-lane halves) ----
  float rmax[8], rsum[8];
#pragma unroll
  for (int r = 0; r < 8; ++r) {
    float mx = s[0][r];
#pragma unroll
    for (int nt = 1; nt < 5; ++nt) mx = fmaxf(mx, s[nt][r]);
#pragma unroll
    for (int d = 1; d <= 8; d <<= 1) mx = fmaxf(mx, __shfl_xor(mx, d, 32));
    rmax[r] = mx;
  }
#pragma unroll
  for (int nt = 0; nt < 5; ++nt)
#pragma unroll
    for (int r = 0; r < 8; ++r) s[nt][r] = __expf(s[nt][r] - rmax[r]);
#pragma unroll
  for (int r = 0; r < 8; ++r) {
    float sm = s[0][r];
#pragma unroll
    for (int nt = 1; nt < 5; ++nt) sm += s[nt][r];
#pragma unroll
    for (int d = 1; d <= 8; d <<= 1) sm += __shfl_xor(sm, d, 32);
    rsum[r] = 1.0f / sm;
  }

  // ---- P to LDS (C-layout -> row-major 16x80) ----
#pragma unroll
  for (int nt = 0; nt < 5; ++nt)
#pragma unroll
    for (int r = 0; r < 8; ++r)
      p_lds[wave][(r + 8 * half) * PSTRIDE + nt * 16 + l16] = s[nt][r] * rsum[r];

  // wave-private LDS write->read ordering (CDNA5 split DS counter)
  asm volatile("s_wait_dscnt 0" ::: "memory");

  // ---- ctx: C(16x64) = P(16x80) * Vwin(80x64) ----
  v8f c[4];
#pragma unroll
  for (int nt = 0; nt < 4; ++nt) c[nt] = zero8();

  const float* pa    = &p_lds[wave][l16 * PSTRIDE + 4 * half];
  const float* vbase = qkv + rowbase * QKV_COLS + 2 * D_MODEL + h * DK;
  for (int k0 = 0; k0 < WKEYS; k0 += 8) {
    v4f a = *(const v4f*)(pa + k0);           // A-frag (b128) from LDS
    int kr = t0 - RADIUS + k0 + 4 * half;     // this lane's four key rows
    const float* v0 = vbase + (size_t)min(max(kr + 0, 0), TLEN - 1) * QKV_COLS;
    const float* v1 = vbase + (size_t)min(max(kr + 1, 0), TLEN - 1) * QKV_COLS;
    const float* v2 = vbase + (size_t)min(max(kr + 2, 0), TLEN - 1) * QKV_COLS;
    const float* v3 = vbase + (size_t)min(max(kr + 3, 0), TLEN - 1) * QKV_COLS;
#pragma unroll
    for (int nt = 0; nt < 4; ++nt) {
      int dk = nt * 16 + l16;
      v2f blo = {v0[dk], v1[dk]};
      v2f bhi = {v2[dk], v3[dk]};
      c[nt] = wmma_f32_k4(lo2(a), blo, c[nt]);
      c[nt] = wmma_f32_k4(hi2(a), bhi, c[nt]);
    }
  }

  // ---- store ctx (ROWS x 512), col = h*64 + dk ----
  float* crow = ctx + (rowbase + t0 + 8 * half) * D_MODEL + h * DK + l16;
#pragma unroll
  for (int nt = 0; nt < 4; ++nt)
#pragma unroll
    for (int r = 0; r < 8; ++r)
      crow[(size_t)r * D_MODEL + nt * 16] = c[nt][r];
}

// ---------------------------------------------------------------------------
extern "C" void kernel_launch(void* const* d_in, const int* in_sizes, int n_in,
                              void* d_out, int out_size, void* d_ws, size_t ws_size,
                              hipStream_t stream) {
  const float* x    = (const float*)d_in[0];
  const float* Wqkv = (const float*)d_in[1];
  const float* bqkv = (const float*)d_in[2];
  const float* Wout = (const float*)d_in[3];
  const float* bout = (const float*)d_in[4];
  float* out = (float*)d_out;

  float* qkv = (float*)d_ws;                          // 8192*1536 f32 = 50.3 MB
  float* ctx = qkv + (size_t)ROWS * QKV_COLS;         // 8192*512  f32 = 16.8 MB

  // QKV projection: (8192x512) x (1536x512)^T
  // 32x64 tiles: (8192/32)*(1536/64) = 6144 wave-tiles, 8 waves/block
  gemm_wmma_f32<<<6144 / 8, 256, 0, stream>>>(
      x, D_MODEL, Wqkv, D_MODEL, bqkv, qkv, QKV_COLS, QKV_COLS / 64, D_MODEL);

  // Banded attention: 4*8*128 = 4096 wave-units
  local_attn_wmma<<<4096 / 8, 256, 0, stream>>>(qkv, ctx);

  // Output projection: (8192x512) x (512x512)^T
  // (8192/32)*(512/64) = 2048 wave-tiles
  gemm_wmma_f32<<<2048 / 8, 256, 0, stream>>>(
      ctx, D_MODEL, Wout, D_MODEL, bout, out, D_MODEL, D_MODEL / 64, D_MODEL);
}